// WeightedSumOfLogits_60644938219545
// MI455X (gfx1250) — compile-verified
//
#include <hip/hip_runtime.h>
#include <math.h>

// ---- problem constants (from reference) ----
#define TOTAL_COLS 20000
#define N_SEG      100
#define BATCH      4096
#define ROWS_PER_TILE 16

typedef __attribute__((ext_vector_type(2))) float v2f;
typedef __attribute__((ext_vector_type(8))) float v8f;

__device__ __forceinline__ int seg_offset(int s) {
    // sizes alternate 150,250 starting with 150 -> offsets always even
    return 150 * ((s + 1) >> 1) + 250 * (s >> 1);
}
__device__ __forceinline__ int seg_len(int s) {
    return (s & 1) ? 250 : 150;        // both are ≡ 2 (mod 4)
}

// ---------------------------------------------------------------------------
// Kernel 1: per-segment softmax of w -> sw (workspace). 100 blocks x 256 thr.
// ---------------------------------------------------------------------------
__global__ void ws_seg_softmax_kernel(const float* __restrict__ w,
                                      float* __restrict__ sw) {
    __shared__ float red[256];
    const int s   = blockIdx.x;
    const int off = seg_offset(s);
    const int len = seg_len(s);          // <= 250 < 256: single pass
    const int t   = threadIdx.x;

    float v = (t < len) ? w[off + t] : -INFINITY;
    red[t] = v;
    __syncthreads();
    #pragma unroll
    for (int st = 128; st > 0; st >>= 1) {
        if (t < st) red[t] = fmaxf(red[t], red[t + st]);
        __syncthreads();
    }
    const float m = red[0];
    __syncthreads();

    const float e = (t < len) ? expf(v - m) : 0.0f;
    red[t] = e;
    __syncthreads();
    #pragma unroll
    for (int st = 128; st > 0; st >>= 1) {
        if (t < st) red[t] += red[t + st];
        __syncthreads();
    }
    const float inv = 1.0f / red[0];
    if (t < len) sw[off + t] = e * inv;
}

// ---------------------------------------------------------------------------
// Kernel 2: streaming weighted segment-sum (HBM-bandwidth bound).
//   grid = (N_SEG, BATCH/16), block = 32 threads (one wave32).
//   Lane map == WMMA f32 16x4 A-matrix layout:
//     lane L: row = L & 15, half = L >> 4; acc pair = K slots {2h, 2h+1}.
//   Hot loop: SGPR-uniform trip count, no exec masking, aligned b64 loads,
//   v_pk_fma_f32, two independent accumulator pairs for ILP.
//   Tail: len % 4 == 2 always -> exactly one extra float2 for half==0 lanes.
//   Epilogue: ONE v_wmma_f32_16x16x4_f32 vs all-ones B reduces the 16x4
//   partial matrix to the 16 row sums (any D column).
// ---------------------------------------------------------------------------
__global__ void ws_weighted_segsum_kernel(const float* __restrict__ A,
                                          const float* __restrict__ sw,
                                          float* __restrict__ out) {
    const int s       = blockIdx.x;
    const int rowBase = blockIdx.y * ROWS_PER_TILE;
    const int off     = seg_offset(s);      // even -> float2 8B-aligned
    const int len     = seg_len(s);

    const int lane = threadIdx.x;           // 0..31
    const int row  = lane & 15;
    const int half = lane >> 4;             // 0 or 1

    const float* __restrict__ pa =
        A + (size_t)(rowBase + row) * TOTAL_COLS + off + 2 * half;
    const float* __restrict__ pw = sw + off + 2 * half;

    const int iters = len >> 2;             // 37 or 62, wave-uniform (SGPR)

    float a0 = 0.0f, a1 = 0.0f;             // accumulator pair 0
    float b0 = 0.0f, b1 = 0.0f;             // accumulator pair 1 (ILP)

    int i = 0;
    #pragma unroll 2
    for (; i + 2 <= iters; i += 2) {
        const v2f av0 = *reinterpret_cast<const v2f*>(pa + 4 * i);
        const v2f av1 = *reinterpret_cast<const v2f*>(pa + 4 * i + 4);
        const v2f wv0 = *reinterpret_cast<const v2f*>(pw + 4 * i);
        const v2f wv1 = *reinterpret_cast<const v2f*>(pw + 4 * i + 4);
        a0 = fmaf(av0.x, wv0.x, a0);
        a1 = fmaf(av0.y, wv0.y, a1);
        b0 = fmaf(av1.x, wv1.x, b0);
        b1 = fmaf(av1.y, wv1.y, b1);
    }
    if (i < iters) {                        // iters odd (150-case): one more
        const v2f av = *reinterpret_cast<const v2f*>(pa + 4 * i);
        const v2f wv = *reinterpret_cast<const v2f*>(pw + 4 * i);
        a0 = fmaf(av.x, wv.x, a0);
        a1 = fmaf(av.y, wv.y, a1);
    }
    // len % 4 == 2: elements [iters*4, iters*4+1] belong to K slots 0,1.
    if (half == 0) {
        const int c = 4 * iters - 0;        // pa already offset by 2*half==0
        const v2f av = *reinterpret_cast<const v2f*>(pa + c);
        const v2f wv = *reinterpret_cast<const v2f*>(pw + c);
        b0 = fmaf(av.x, wv.x, b0);
        b1 = fmaf(av.y, wv.y, b1);
    }

    // Merge ILP pairs -> exact WMMA f32 16x4 A-matrix layout (2 VGPRs/lane).
    v2f amat; amat.x = a0 + b0; amat.y = a1 + b1;
    v2f bmat; bmat.x = 1.0f;   bmat.y = 1.0f;    // all-ones B
    v8f cacc = {};
    cacc = __builtin_amdgcn_wmma_f32_16x16x4_f32(
        /*neg_a=*/false, amat, /*neg_b=*/false, bmat,
        /*c_mod=*/(short)0, cacc, /*reuse_a=*/false, /*reuse_b=*/false);

    // D layout: VGPR v, lanes 0-15 -> M=v, lanes 16-31 -> M=v+8; N = lane%16.
    if ((lane & 15) == 0) {
        const int mbase = half * 8;
        float* __restrict__ orow = out + (size_t)(rowBase + mbase) * N_SEG + s;
        #pragma unroll
        for (int m = 0; m < 8; ++m) {
            orow[(size_t)m * N_SEG] = cacc[m];
        }
    }
}

// ---------------------------------------------------------------------------
extern "C" void kernel_launch(void* const* d_in, const int* in_sizes, int n_in,
                              void* d_out, int out_size, void* d_ws, size_t ws_size,
                              hipStream_t stream) {
    (void)in_sizes; (void)n_in; (void)out_size; (void)ws_size;
    const float* A = (const float*)d_in[0];   // [4096, 20000] f32
    const float* w = (const float*)d_in[1];   // [20000] f32
    float* out = (float*)d_out;               // [4096, 100] f32
    float* sw  = (float*)d_ws;                // [20000] f32 scratch

    hipLaunchKernelGGL(ws_seg_softmax_kernel, dim3(N_SEG), dim3(256), 0, stream,
                       w, sw);
    hipLaunchKernelGGL(ws_weighted_segsum_kernel,
                       dim3(N_SEG, BATCH / ROWS_PER_TILE), dim3(32), 0, stream,
                       A, sw, out);
}